// MLS_76115410420211
// MI455X (gfx1250) — compile-verified
//
#include <hip/hip_runtime.h>

// MLS affine deformation + gather for MI455X (gfx1250, wave32).
//
// Phase 1 (per wave, 16 pixels): the 12 weighted point-moments needed by the
// MLS affine solve are accumulated on the matrix pipe as
//   D[16 moments x 16 pixels] += A[16 moments x 4 points] * B[4 points x 16 pixels]
// with 8 chained V_WMMA_F32_16X16X4_F32 (K=32 control points). VALU only
// computes the 32 reciprocal weights (v_rcp_f32) and the 2x2 epilogue, and
// co-executes with the XDL WMMA pipe.
// Phase 2: memory-bound gather, coalesced 128-float stores per channel row,
// with global_prefetch_b8 for the next channel's gather line.

typedef __attribute__((ext_vector_type(2))) float v2f;
typedef __attribute__((ext_vector_type(8))) float v8f;

#define HH   512
#define WW   512
#define CCH  64
#define NP   32          // control points
#define EPSF 1e-8f

__global__ __launch_bounds__(256) void mls_wmma_gather(
    const float* __restrict__ fea,   // [B, C, H, W]
    const int*   __restrict__ Ld,    // [B, 2, NP]  (q source points)
    const int*   __restrict__ Lg,    // [B, 2, NP]  (p dest points)
    float*       __restrict__ out)   // [B, C, H, W]
{
    // F[pt][m]: per-point moment features (A-matrix source), 2 KB.
    __shared__ float F[NP * 16];
    __shared__ int   gidx[128];      // gather offsets (row*W+col) for 128 pixels

    const int t = threadIdx.x;
    const int b = blockIdx.y;

    // ---- Build the per-batch moment table (32 threads, trivial) ----
    if (t < NP) {
        const int pt = t;
        const float p0 = (float)Lg[b * 2 * NP + pt];
        const float p1 = (float)Lg[b * 2 * NP + NP + pt];
        const float q0 = (float)Ld[b * 2 * NP + pt];
        const float q1 = (float)Ld[b * 2 * NP + NP + pt];
        float* f = &F[pt * 16];
        f[0]  = 1.0f;            // -> sum(w) for normalization
        f[1]  = p0;  f[2]  = p1; // -> pstar
        f[3]  = q0;  f[4]  = q1; // -> qstar
        f[5]  = p0 * p0; f[6] = p0 * p1; f[7] = p1 * p1;   // -> pTwp
        f[8]  = p0 * q0; f[9] = p0 * q1;                   // -> M = sum w p q^T
        f[10] = p1 * q0; f[11] = p1 * q1;
        f[12] = 0.0f; f[13] = 0.0f; f[14] = 0.0f; f[15] = 0.0f;
    }
    __syncthreads();

    const int lane = t & 31;
    const int wave = t >> 5;
    const int P    = lane & 15;   // pixel slot (N) == moment slot (M) for A
    const int hi   = lane >> 4;   // lane-half selects K pair {k, k+1} vs {k+2, k+3}

    const int pixBase = blockIdx.x * 128;
    const int pix     = pixBase + wave * 16 + P;
    const float vx    = (float)(pix >> 9);        // row   (H = 512)
    const float vy    = (float)(pix & (WW - 1));  // col

    // ---- Moment accumulation on the matrix pipe: 8 x wmma_f32_16x16x4_f32 ----
    v8f acc = {0.f, 0.f, 0.f, 0.f, 0.f, 0.f, 0.f, 0.f};
#pragma unroll
    for (int j = 0; j < 8; ++j) {
        const int k0 = 4 * j + 2 * hi;   // this lane's two K (point) indices
        const int k1 = k0 + 1;

        // A operand (16x4, M=lane&15, VGPR0/1 = K pair): point moment features
        const float a0 = F[k0 * 16 + P];
        const float a1 = F[k1 * 16 + P];

        // B operand (4x16, N=lane&15, lane-half splits K): MLS weights
        const float dx0 = vx - F[k0 * 16 + 1];
        const float dy0 = vy - F[k0 * 16 + 2];
        const float w0  = __builtin_amdgcn_rcpf(dx0 * dx0 + dy0 * dy0 + EPSF);
        const float dx1 = vx - F[k1 * 16 + 1];
        const float dy1 = vy - F[k1 * 16 + 2];
        const float w1  = __builtin_amdgcn_rcpf(dx1 * dx1 + dy1 * dy1 + EPSF);

        v2f A  = {a0, a1};
        v2f Bv = {w0, w1};
        // (neg_a, A, neg_b, B, c_mod, C, reuse_a, reuse_b)
        acc = __builtin_amdgcn_wmma_f32_16x16x4_f32(
            false, A, false, Bv, (short)0, acc, false, false);
    }

    // D layout: lane&15 = pixel; VGPR g = moment g (lanes 0-15) / 8+g (lanes 16-31).
    // Pull moments 8..11 across to the low lanes.
    const float m8  = __shfl(acc[0], P + 16, 32);
    const float m9  = __shfl(acc[1], P + 16, 32);
    const float m10 = __shfl(acc[2], P + 16, 32);
    const float m11 = __shfl(acc[3], P + 16, 32);

    if (lane < 16) {
        const float rsw = 1.0f / acc[0];              // 1 / sum(w) -> normalize
        const float p0  = acc[1] * rsw, p1 = acc[2] * rsw;   // pstar
        const float q0  = acc[3] * rsw, q1 = acc[4] * rsw;   // qstar
        const float s00 = acc[5] * rsw, s01 = acc[6] * rsw, s11 = acc[7] * rsw;
        const float t00 = m8 * rsw,  t01 = m9 * rsw;
        const float t10 = m10 * rsw, t11 = m11 * rsw;

        // pTwp = S_pp - pstar pstar^T   (symmetric 2x2)
        const float a  = s00 - p0 * p0;
        const float bb = s01 - p0 * p1;
        const float dd = s11 - p1 * p1;
        const float rdet = 1.0f / (a * dd - bb * bb);

        // tmp = (v - pstar) * inv(pTwp)
        const float ml0 = vx - p0, ml1 = vy - p1;
        const float u0 = (ml0 * dd - ml1 * bb) * rdet;
        const float u1 = (ml1 * a  - ml0 * bb) * rdet;

        // M = S_pq - pstar qstar^T ;  trans = tmp * M + qstar
        const float M00 = t00 - p0 * q0, M01 = t01 - p0 * q1;
        const float M10 = t10 - p1 * q0, M11 = t11 - p1 * q1;
        float tr0 = u0 * M00 + u1 * M10 + q0;
        float tr1 = u0 * M01 + u1 * M11 + q1;

        // Reference clamp semantics: <0 -> 0 ; > (dim-1) -> 0 ; truncate to int
        tr0 = (tr0 < 0.f) ? 0.f : tr0;
        tr1 = (tr1 < 0.f) ? 0.f : tr1;
        tr0 = (tr0 > (float)(HH - 1)) ? 0.f : tr0;
        tr1 = (tr1 > (float)(WW - 1)) ? 0.f : tr1;
        gidx[wave * 16 + P] = (int)tr0 * WW + (int)tr1;
    }
    __syncthreads();

    // ---- Memory-bound gather: 64 channels x 128 pixels per block ----
    const size_t planes = (size_t)CCH * HH * WW;
    const float* feab = fea + (size_t)b * planes;
    float*       outb = out + (size_t)b * planes;
#pragma unroll 4
    for (int i = t; i < CCH * 128; i += 256) {
        const int c  = i >> 7;
        const int pp = i & 127;
        const float* src = feab + c * (HH * WW) + gidx[pp];
        if (c + 2 < CCH)
            __builtin_prefetch(src + 2 * (HH * WW), 0, 0);  // global_prefetch_b8
        outb[c * (HH * WW) + pixBase + pp] = *src;
    }
}

extern "C" void kernel_launch(void* const* d_in, const int* in_sizes, int n_in,
                              void* d_out, int out_size, void* d_ws, size_t ws_size,
                              hipStream_t stream) {
    const float* fea = (const float*)d_in[0];   // fea_g  [B,C,H,W] f32
    const int*   Ld  = (const int*)d_in[1];     // L_d    [B,2,32]  i32
    const int*   Lg  = (const int*)d_in[2];     // L_g    [B,2,32]  i32
    float* out = (float*)d_out;

    const int B = in_sizes[1] / (2 * NP);       // 4
    dim3 grid((HH * WW) / 128, B, 1);           // 2048 x B blocks, 128 px each
    mls_wmma_gather<<<grid, 256, 0, stream>>>(fea, Ld, Lg, out);
}